// LatentSensorModel_20358144983236
// MI455X (gfx1250) — compile-verified
//
#include <hip/hip_runtime.h>
#include <hip/hip_bf16.h>
#include <math.h>

// ---------------------------------------------------------------------------
// LatentSensorModel: per-row 2-component 2D-Gaussian mixture log-likelihood.
// Bandwidth-bound (16 B/row -> ~134 MB total -> ~5.8 us floor at 23.3 TB/s).
// Per-source params precomputed into a 48 KB table, staged into LDS per block
// via CDNA5 async global->LDS (ASYNCcnt / s_wait_asynccnt), data streamed as
// b128 vectors with non-temporal hints.
// ---------------------------------------------------------------------------

#define LOG_2PI_F 1.8378770664093453f
#define MAX_S 1024
#define TAB_STRIDE 12                    // floats per source (48 B, 16B-aligned)
#define TAB_FLOATS (MAX_S * TAB_STRIDE)  // 12288 floats = 48 KB
#define BLOCK 256
#define GRID_BLOCKS 1024

typedef float __attribute__((ext_vector_type(4))) fvec4;
typedef float __attribute__((ext_vector_type(2))) fvec2;
typedef int   __attribute__((ext_vector_type(4))) ivec4;

// GCC-style vector for the async builtin's typed pointer params (v4i AS1* / AS3*)
typedef int v4i_gcc __attribute__((vector_size(16)));
typedef __attribute__((address_space(1))) v4i_gcc* g_v4i_ptr;
typedef __attribute__((address_space(3))) v4i_gcc* l_v4i_ptr;

// ---- CDNA5 async global->LDS copy (b128 per lane) -------------------------
__device__ __forceinline__ void async_b128_to_lds(const void* gsrc, void* ldst) {
#if __has_builtin(__builtin_amdgcn_global_load_async_to_lds_b128)
  __builtin_amdgcn_global_load_async_to_lds_b128(
      (g_v4i_ptr)(unsigned long long)gsrc,
      (l_v4i_ptr)(unsigned)(unsigned long long)ldst,
      0, 0);
#else
  unsigned lds_off = (unsigned)(unsigned long long)ldst;  // flat LDS aperture: low 32 bits = LDS byte offset
  asm volatile("global_load_async_to_lds_b128 %0, %1, off"
               :
               : "v"(lds_off), "v"(gsrc)
               : "memory");
#endif
}

__device__ __forceinline__ void wait_asynccnt0() {
#if __has_builtin(__builtin_amdgcn_s_wait_asynccnt)
  __builtin_amdgcn_s_wait_asynccnt(0);
#else
  asm volatile("s_wait_asynccnt 0" ::: "memory");
#endif
}

// ---- Kernel 1: fold all per-source math into a 12-float record ------------
// layout per source s (48 B):
//  [0]=m_u [1]=m_v [2]=base0 [3]=base1 [4]=A0 [5]=B0 [6]=D0 [7]=A1 [8]=B1 [9]=D1 [10..11]=pad
// l_k(dx,dy) = base_k + A_k*dx*dx + B_k*dx*dy + D_k*dy*dy
__global__ void lsm_prep_kernel(const float* __restrict__ mu_u,
                                const float* __restrict__ log_var_u,
                                const float* __restrict__ a,
                                const float* __restrict__ b,
                                const float* __restrict__ log_var_v,
                                const float* __restrict__ raw_pi,
                                const float* __restrict__ raw_sigma0,
                                const float* __restrict__ raw_delta_sigma,
                                float* __restrict__ tab, int S) {
  int s = blockIdx.x * blockDim.x + threadIdx.x;
  if (s >= S) return;

  // softplus(x) = max(x,0) + log1p(exp(-|x|))  (matches jax.nn.softplus)
  float r0 = raw_sigma0[0];
  float r1 = raw_delta_sigma[0];
  float sp0 = fmaxf(r0, 0.0f) + log1pf(expf(-fabsf(r0)));
  float sp1 = fmaxf(r1, 0.0f) + log1pf(expf(-fabsf(r1)));
  float sig0 = sp0 + 1e-6f;
  float sig1 = sig0 + sp1;
  float s0sq = sig0 * sig0 + 1e-12f;
  float s1sq = sig1 * sig1 + 1e-12f;

  float vu = fmaxf(expf(log_var_u[s]), 1e-9f);
  float vv = fmaxf(expf(log_var_v[s]), 1e-9f);
  float ar = a[s];
  float mu = mu_u[s];
  float mv = ar * mu + b[s];
  float P00 = vu;
  float P01 = ar * vu;
  float P11 = ar * ar * vu + vv;
  float pi = 1.0f / (1.0f + expf(-raw_pi[s]));
  float w0 = logf(pi + 1e-12f);
  float w1 = logf(1.0f - pi + 1e-12f);

  float rec[TAB_STRIDE];
  rec[0] = mu;
  rec[1] = mv;
  {
    float c00 = P00 + s0sq, c01 = P01, c11 = P11 + s0sq;
    float det = c00 * c11 - c01 * c01;
    float inv = 1.0f / det;
    rec[2] = w0 - 0.5f * (2.0f * LOG_2PI_F + logf(det));
    rec[4] = -0.5f * c11 * inv;
    rec[5] = c01 * inv;
    rec[6] = -0.5f * c00 * inv;
  }
  {
    float c00 = P00 + s1sq, c01 = P01, c11 = P11 + s1sq;
    float det = c00 * c11 - c01 * c01;
    float inv = 1.0f / det;
    rec[3] = w1 - 0.5f * (2.0f * LOG_2PI_F + logf(det));
    rec[7] = -0.5f * c11 * inv;
    rec[8] = c01 * inv;
    rec[9] = -0.5f * c00 * inv;
  }
  rec[10] = 0.0f;
  rec[11] = 0.0f;

#pragma unroll
  for (int i = 0; i < TAB_STRIDE; ++i) tab[s * TAB_STRIDE + i] = rec[i];
}

// ---- per-element evaluation against the LDS table -------------------------
__device__ __forceinline__ float lsm_eval(const float* __restrict__ p, float u, float d) {
  fvec4 h0 = *(const fvec4*)(p);      // m_u, m_v, base0, base1
  fvec4 h1 = *(const fvec4*)(p + 4);  // A0, B0, D0, A1
  fvec2 h2 = *(const fvec2*)(p + 8);  // B1, D1
  float dx = u - h0[0];
  float dy = d - h0[1];
  float dxx = dx * dx, dxy = dx * dy, dyy = dy * dy;
  float l0 = h0[2] + h1[0] * dxx + h1[1] * dxy + h1[2] * dyy;
  float l1 = h0[3] + h1[3] * dxx + h2[0] * dxy + h2[1] * dyy;
  float m = fmaxf(l0, l1);
  float t = __expf(-fabsf(l0 - l1));  // in (0,1], well conditioned
  return m + log1pf(t);
}

// ---- Kernel 2: stream N rows, gather from LDS table ------------------------
__global__ __launch_bounds__(BLOCK) void lsm_main_kernel(const int* __restrict__ src,
                                                         const float* __restrict__ up,
                                                         const float* __restrict__ dn,
                                                         const float* __restrict__ tab_g,
                                                         float* __restrict__ out,
                                                         int n, int S) {
  __shared__ fvec4 tab4[TAB_FLOATS / 4];
  float* tab = (float*)tab4;

  // Stage the parameter table into LDS with the async copy engine.
  int nvec = (S < MAX_S ? S : MAX_S) * TAB_STRIDE / 4;  // 3072 b128 transfers for S=1024
  const fvec4* g4 = (const fvec4*)tab_g;
  for (int i = threadIdx.x; i < nvec; i += BLOCK) {
    async_b128_to_lds(&g4[i], &tab4[i]);
  }
  wait_asynccnt0();
  __syncthreads();

  int nquad = n >> 2;
  const ivec4* src4 = (const ivec4*)src;
  const fvec4* up4  = (const fvec4*)up;
  const fvec4* dn4  = (const fvec4*)dn;
  fvec4* out4       = (fvec4*)out;

  int stride = gridDim.x * BLOCK;
  for (int q = blockIdx.x * BLOCK + threadIdx.x; q < nquad; q += stride) {
    ivec4 s4 = __builtin_nontemporal_load(&src4[q]);
    fvec4 u4 = __builtin_nontemporal_load(&up4[q]);
    fvec4 d4 = __builtin_nontemporal_load(&dn4[q]);
    fvec4 o;
#pragma unroll
    for (int c = 0; c < 4; ++c) {
      const float* p = &tab[s4[c] * TAB_STRIDE];
      o[c] = lsm_eval(p, u4[c], d4[c]);
    }
    __builtin_nontemporal_store(o, &out4[q]);
  }

  // Scalar tail (N % 4), handled by block 0 out of the already-loaded LDS table.
  int rem = n - (nquad << 2);
  if (blockIdx.x == 0 && (int)threadIdx.x < rem) {
    int i = (nquad << 2) + threadIdx.x;
    const float* p = &tab[src[i] * TAB_STRIDE];
    out[i] = lsm_eval(p, up[i], dn[i]);
  }
}

// ---------------------------------------------------------------------------
extern "C" void kernel_launch(void* const* d_in, const int* in_sizes, int n_in,
                              void* d_out, int out_size, void* d_ws, size_t ws_size,
                              hipStream_t stream) {
  const int*   source    = (const int*)d_in[0];
  const float* upstream  = (const float*)d_in[1];
  const float* downstream= (const float*)d_in[2];
  const float* mu_u      = (const float*)d_in[3];
  const float* log_var_u = (const float*)d_in[4];
  const float* a         = (const float*)d_in[5];
  const float* b         = (const float*)d_in[6];
  const float* log_var_v = (const float*)d_in[7];
  const float* raw_pi    = (const float*)d_in[8];
  const float* raw_sigma0      = (const float*)d_in[9];
  const float* raw_delta_sigma = (const float*)d_in[10];

  int n = in_sizes[0];
  int S = in_sizes[3];
  if (S > MAX_S) S = MAX_S;

  float* tab = (float*)d_ws;  // S * 12 floats = 48 KB for S=1024

  lsm_prep_kernel<<<(S + BLOCK - 1) / BLOCK, BLOCK, 0, stream>>>(
      mu_u, log_var_u, a, b, log_var_v, raw_pi, raw_sigma0, raw_delta_sigma, tab, S);

  int nquad = n >> 2;
  int blocks = (nquad + BLOCK - 1) / BLOCK;
  if (blocks > GRID_BLOCKS) blocks = GRID_BLOCKS;
  if (blocks < 1) blocks = 1;

  lsm_main_kernel<<<blocks, BLOCK, 0, stream>>>(
      source, upstream, downstream, tab, (float*)d_out, n, S);
}